// ScratchMPNNLayer_72232759984909
// MI455X (gfx1250) — compile-verified
//
#include <hip/hip_runtime.h>
#include <hip/hip_bf16.h>

#define NN 100000
#define EE 1600000
#define DIN 64
#define DOUT 64

typedef __attribute__((ext_vector_type(2))) float v2f;
typedef __attribute__((ext_vector_type(4))) float v4f;
typedef __attribute__((ext_vector_type(8))) float v8f;

// ---------------------------------------------------------------------------
// K0: zero agg accumulator (N*64) and degree (N)
// ---------------------------------------------------------------------------
__global__ void __launch_bounds__(256) zero_ws_kernel(float* agg, float* deg) {
    const long long total = (long long)NN * 64 + NN;
    long long i = (long long)blockIdx.x * blockDim.x + threadIdx.x;
    if (i >= total) return;
    if (i < (long long)NN * 64) agg[i] = 0.0f;
    else                        deg[i - (long long)NN * 64] = 0.0f;
}

// ---------------------------------------------------------------------------
// K1: fold weights:  W_comb = W_msg @ W_upd[64:128,:]   (64x64)
//                    c_vec  = b_msg @ W_upd[64:128,:]   (64)
// ---------------------------------------------------------------------------
__global__ void __launch_bounds__(256) fold_weights_kernel(
    const float* __restrict__ Wmsg, const float* __restrict__ bmsg,
    const float* __restrict__ Wupd, float* __restrict__ Wcomb,
    float* __restrict__ cvec) {
    int idx = blockIdx.x * 256 + threadIdx.x;
    if (idx < 64 * 64) {
        int i = idx >> 6, j = idx & 63;
        float s = 0.0f;
        #pragma unroll 8
        for (int k = 0; k < 64; ++k)
            s += Wmsg[i * 64 + k] * Wupd[(64 + k) * 64 + j];
        Wcomb[idx] = s;
    } else if (idx < 64 * 64 + 64) {
        int j = idx - 64 * 64;
        float s = 0.0f;
        #pragma unroll 8
        for (int k = 0; k < 64; ++k)
            s += bmsg[k] * Wupd[(64 + k) * 64 + j];
        cvec[j] = s;
    }
}

// ---------------------------------------------------------------------------
// K2: edge scatter.  16 lanes per edge; each lane moves 4 contiguous floats
// (one b128 load of the source row slice, four global_atomic_add_f32).
// agg (25.6MB) and deg are L2-resident -> atomics stay in L2.
// ---------------------------------------------------------------------------
__global__ void __launch_bounds__(256) edge_scatter_kernel(
    const float* __restrict__ x, const int* __restrict__ edge_index,
    float* __restrict__ agg, float* __restrict__ deg) {
    long long t = (long long)blockIdx.x * 256 + threadIdx.x;
    long long e = t >> 4;
    if (e >= EE) return;
    int l4  = (int)(t & 15);
    int dst = edge_index[e];        // row  (destination, segment id)
    int src = edge_index[EE + e];   // col  (source node features)
    const float4 v = *(const float4*)(x + (size_t)src * 64 + l4 * 4);
    float* p = agg + (size_t)dst * 64 + l4 * 4;
    atomicAdd(p + 0, v.x);
    atomicAdd(p + 1, v.y);
    atomicAdd(p + 2, v.z);
    atomicAdd(p + 3, v.w);
    if (l4 == 0) atomicAdd(deg + dst, 1.0f);
}

// ---------------------------------------------------------------------------
// K3: node phase, fp32 WMMA (V_WMMA_F32_16X16X4_F32).
// out = relu( x @ Wtop + agg_x @ Wcomb + deg (x) c_vec + b_upd )
// One wave per 16-row tile; 8 waves per block.
//
// LDS weight layout = exact B-fragment shape: for k-pair kk = k>>1 and
// column col, sWB[(kk*64+col)*4 + {0,1,2,3}] = {Wtop[2kk][col], Wtop[2kk+1][col],
// Wcomb[2kk][col], Wcomb[2kk+1][col]}.  One 16B-aligned ds_load_b128 per
// k-step yields BOTH B fragments in adjacent even-aligned VGPR pairs.
//
// A fragment layout (16x4 f32): lane = M + 16*(K/2), vgpr = K%2.
// B fragment layout (4x16 f32): lane = N + 16*(K/2), vgpr = K%2.
// C/D layout: vgpr v -> M = v + 8*(lane/16), N = lane%16.
// ---------------------------------------------------------------------------
__global__ void __launch_bounds__(256) node_wmma_kernel(
    const float* __restrict__ x, const float* __restrict__ agg,
    const float* __restrict__ deg, const float* __restrict__ Wupd,
    const float* __restrict__ bupd, const float* __restrict__ Wcomb,
    const float* __restrict__ cvec, float* __restrict__ out, int nTiles) {
    __shared__ __align__(16) float sWB[32 * 64 * 4];  // 32KB interleaved weights
    __shared__ float sCvec[64];
    __shared__ float sBupd[64];

    // Cooperative fill into B-fragment layout.
    for (int i = threadIdx.x; i < 64 * 64; i += 256) {
        int k = i >> 6, col = i & 63;
        int base = (((k >> 1) * 64 + col) << 2) + (k & 1);
        sWB[base]     = Wupd[i];    // Wtop  slot {0,1}
        sWB[base + 2] = Wcomb[i];   // Wcomb slot {2,3}
    }
    if (threadIdx.x < 64) {
        sCvec[threadIdx.x] = cvec[threadIdx.x];
        sBupd[threadIdx.x] = bupd[threadIdx.x];
    }
    __syncthreads();

    const int wave = threadIdx.x >> 5;
    const int lane = threadIdx.x & 31;
    const int tile = blockIdx.x * 8 + wave;
    if (tile >= nTiles) return;              // wave-uniform: EXEC stays all-1s

    const int r0   = tile * 16;
    const int m    = lane & 15;              // A row / B,C col index base
    const int half = lane >> 4;              // 0: K%4 in {0,1}, 1: K%4 in {2,3}
    const int kp   = half * 2;

    // Preload both A tiles (16x64 each) into registers in WMMA A-layout.
    v2f ax[16], ag[16];
    const float* xrow = x   + (size_t)(r0 + m) * 64 + kp;
    const float* grow = agg + (size_t)(r0 + m) * 64 + kp;
    #pragma unroll
    for (int i = 0; i < 16; ++i) {
        ax[i] = *(const v2f*)(xrow + 4 * i);
        ag[i] = *(const v2f*)(grow + 4 * i);
    }
    // Per-lane degrees for the 8 rows this lane-half owns in C/D.
    float degv[8];
    #pragma unroll
    for (int v = 0; v < 8; ++v) degv[v] = deg[r0 + half * 8 + v];

    #pragma unroll
    for (int nt = 0; nt < 4; ++nt) {         // 4 column tiles of 16
        const int col = nt * 16 + m;
        v8f acc = {};
        #pragma unroll
        for (int i = 0; i < 16; ++i) {       // K = 64 in steps of 4
            const int kk = 2 * i + half;     // k-pair index for this lane half
            const v4f w = *(const v4f*)&sWB[(kk * 64 + col) << 2];
            v2f bt = { w.x, w.y };           // Wtop  fragment
            v2f bc = { w.z, w.w };           // Wcomb fragment
            acc = __builtin_amdgcn_wmma_f32_16x16x4_f32(
                false, ax[i], false, bt, (short)0, acc, false, false);
            acc = __builtin_amdgcn_wmma_f32_16x16x4_f32(
                false, ag[i], false, bc, (short)0, acc, false, false);
        }
        const float cv = sCvec[col];
        const float bu = sBupd[col];
        #pragma unroll
        for (int v = 0; v < 8; ++v) {
            const int mm = half * 8 + v;
            float val = acc[v] + degv[v] * cv + bu;
            out[(size_t)(r0 + mm) * 64 + col] = fmaxf(val, 0.0f);
        }
    }
}

// ---------------------------------------------------------------------------
extern "C" void kernel_launch(void* const* d_in, const int* in_sizes, int n_in,
                              void* d_out, int out_size, void* d_ws, size_t ws_size,
                              hipStream_t stream) {
    const float* x          = (const float*)d_in[0];
    const int*   edge_index = (const int*)d_in[1];
    const float* W_msg      = (const float*)d_in[2];
    const float* b_msg      = (const float*)d_in[3];
    const float* W_upd      = (const float*)d_in[4];
    const float* b_upd      = (const float*)d_in[5];
    float* out = (float*)d_out;

    // workspace layout (floats): agg[N*64] | deg[N] | Wcomb[4096] | cvec[64]
    float* agg   = (float*)d_ws;
    float* deg   = agg + (size_t)NN * 64;
    float* Wcomb = deg + NN;
    float* cvec  = Wcomb + 64 * 64;

    {   // K0: zero accumulators
        long long total = (long long)NN * 64 + NN;
        int blocks = (int)((total + 255) / 256);
        zero_ws_kernel<<<blocks, 256, 0, stream>>>(agg, deg);
    }
    {   // K1: fold weights (tiny)
        fold_weights_kernel<<<17, 256, 0, stream>>>(W_msg, b_msg, W_upd, Wcomb, cvec);
    }
    {   // K2: edge scatter, 16 lanes per edge
        long long threads = (long long)EE * 16;
        int blocks = (int)((threads + 255) / 256);
        edge_scatter_kernel<<<blocks, 256, 0, stream>>>(x, edge_index, agg, deg);
    }
    {   // K3: WMMA node phase
        const int nTiles = (NN + 15) / 16;           // 6250
        int blocks = (nTiles + 7) / 8;               // 8 waves/block
        node_wmma_kernel<<<blocks, 256, 0, stream>>>(x, agg, deg, W_upd, b_upd,
                                                     Wcomb, cvec, out, nTiles);
    }
}